// DynoNet_40424232190449
// MI455X (gfx1250) — compile-verified
//
#include <hip/hip_runtime.h>

typedef __attribute__((ext_vector_type(2))) float v2f;
typedef __attribute__((ext_vector_type(8))) float v8f;

namespace {
constexpr int BATCH = 16;
constexpr int T     = 8192;
constexpr int CIN   = 4;
constexpr int ENC   = 32;
constexpr int HID   = 128;
constexpr int OUTC  = 4;
constexpr int NA    = 4;
constexpr int NBT   = 9;               // N_B + 1 taps
constexpr int ROWS  = BATCH * T;       // 131072
constexpr int NTIL  = ROWS / 16;       // 8192 row tiles (never cross a batch: 8192 % 16 == 0)
}

static __device__ __forceinline__ v8f wmma_k4(v2f a, v2f b, v8f c) {
  // D(16x16,f32) = A(16x4,f32) * B(4x16,f32) + C  -> v_wmma_f32_16x16x4_f32
  return __builtin_amdgcn_wmma_f32_16x16x4_f32(false, a, false, b, (short)0, c, false, false);
}

static __device__ __forceinline__ void wait_ds0() {
  asm volatile("s_wait_dscnt 0" ::: "memory");
}

// ---------------------------------------------------------------------------
// Kernel 1: FIR1  bu1[b,t,e] = sum_{tap,c} u[b,t-tap,c] * b1[tap,c,e]
// K = 9 taps * 4 cin = 36 -> 9 WMMA K-steps; N = 32 -> 2 N-tiles.
// B-fragments are tile-invariant: hoisted to registers before the tile loop.
// ---------------------------------------------------------------------------
__global__ void __launch_bounds__(256) k_fir1(const float* __restrict__ u,
                                              const float* __restrict__ b1,
                                              float* __restrict__ bu) {
  __shared__ float sB[NBT * CIN * ENC];  // 1152 floats
  for (int i = threadIdx.x; i < NBT * CIN * ENC; i += blockDim.x) sB[i] = b1[i];
  __syncthreads();

  const int lane = threadIdx.x & 31, ln = lane & 15, hi = lane >> 4;
  const int wid = blockIdx.x * (blockDim.x >> 5) + (threadIdx.x >> 5);
  const int nw  = gridDim.x * (blockDim.x >> 5);

  // Hoisted, tile-invariant B fragments (9 taps x 2 N-tiles).
  v2f bf0[NBT], bf1[NBT];
#pragma unroll
  for (int tap = 0; tap < NBT; ++tap) {
    const int k = tap * CIN + 2 * hi;
    bf0[tap].x = sB[(k    ) * ENC +      ln];
    bf0[tap].y = sB[(k + 1) * ENC +      ln];
    bf1[tap].x = sB[(k    ) * ENC + 16 + ln];
    bf1[tap].y = sB[(k + 1) * ENC + 16 + ln];
  }
  wait_ds0();

  for (int tile = wid; tile < NTIL; tile += nw) {
    const int row0 = tile * 16;
    const int b = row0 / T, t0 = row0 % T;

    v2f af[NBT];
    if (t0 != 0) {
      // Interior tile: single base, immediate (negative) offsets.
      const float* ub = u + ((size_t)b * T + t0 + ln) * CIN + 2 * hi;
#pragma unroll
      for (int tap = 0; tap < NBT; ++tap)
        af[tap] = *(const v2f*)(ub - tap * CIN);
    } else {
      // Boundary tile (t0 == 0): zero-pad the causal window.
#pragma unroll
      for (int tap = 0; tap < NBT; ++tap) {
        const int t = ln - tap;
        v2f v = {};
        if (t >= 0) v = *(const v2f*)(u + ((size_t)b * T + t) * CIN + 2 * hi);
        af[tap] = v;
      }
    }

    v8f acc0 = {}, acc1 = {};
#pragma unroll
    for (int tap = 0; tap < NBT; ++tap) {
      acc0 = wmma_k4(af[tap], bf0[tap], acc0);
      acc1 = wmma_k4(af[tap], bf1[tap], acc1);
    }

    // D tile store: one base pointer, constant row offsets (j*128 bytes).
    float* p = bu + (size_t)(row0 + 8 * hi) * ENC + ln;
#pragma unroll
    for (int j = 0; j < 8; ++j) {
      p[j * ENC]      = acc0[j];
      p[j * ENC + 16] = acc1[j];
    }
  }
}

// ---------------------------------------------------------------------------
// Kernel 2: IIR1 in-place over bu1: y[t] = v[t] + sum_n a[n]*y[t-1-n], x0 = 0
// One chain per lane: 16*32 = 512 lanes.
// ---------------------------------------------------------------------------
__global__ void k_iir_enc(float* __restrict__ buf, const float* __restrict__ a) {
  const int tid = blockIdx.x * blockDim.x + threadIdx.x;
  if (tid >= BATCH * ENC) return;
  const int b = tid / ENC, e = tid % ENC;
  const float a0 = a[0 * ENC + e], a1c = a[1 * ENC + e];
  const float a2c = a[2 * ENC + e], a3c = a[3 * ENC + e];
  float y1 = 0.f, y2 = 0.f, y3 = 0.f, y4 = 0.f;
  float* p = buf + (size_t)b * T * ENC + e;
  for (int t = 0; t < T; ++t) {
    float v = *p;
    v = fmaf(a0, y1, fmaf(a1c, y2, fmaf(a2c, y3, fmaf(a3c, y4, v))));
    *p = v;
    p += ENC;
    y4 = y3; y3 = y2; y2 = y1; y1 = v;
  }
}

// ---------------------------------------------------------------------------
// Kernel 3: MLP (in-place buf: x -> xf). GEMM1 [16x32]*[32x128] + sigmoid,
// per-wave LDS transpose of h, GEMM2 [16x128]*[128x32]. 4 waves / block.
// ---------------------------------------------------------------------------
__global__ void __launch_bounds__(128) k_mlp(float* __restrict__ buf,
                                             const float* __restrict__ W1,
                                             const float* __restrict__ c1,
                                             const float* __restrict__ W2,
                                             const float* __restrict__ c2) {
  __shared__ float sW1[ENC * HID];     // 16 KB
  __shared__ float sW2[HID * ENC];     // 16 KB
  __shared__ float sH[4][16 * HID];    // 8 KB per wave
  for (int i = threadIdx.x; i < ENC * HID; i += blockDim.x) sW1[i] = W1[i];
  for (int i = threadIdx.x; i < HID * ENC; i += blockDim.x) sW2[i] = W2[i];
  __syncthreads();

  const int lane = threadIdx.x & 31, ln = lane & 15, hi = lane >> 4;
  const int wv  = threadIdx.x >> 5;
  const int wid = blockIdx.x * 4 + wv;
  const int nw  = gridDim.x * 4;
  float* hb = &sH[wv][0];

  float c1f[8], c2f[2];
#pragma unroll
  for (int nt = 0; nt < 8; ++nt) c1f[nt] = c1[nt * 16 + ln];
  c2f[0] = c2[ln]; c2f[1] = c2[16 + ln];

  for (int tile = wid; tile < NTIL; tile += nw) {
    const int row0 = tile * 16;
    if (tile + nw < NTIL)
      __builtin_prefetch(buf + (size_t)(tile + nw) * 16 * ENC, 0, 0);

    // ---- GEMM1: A = x tile (16x32), B = W1 (32x128) ----
    const float* xr = buf + (size_t)(row0 + ln) * ENC + 2 * hi;
    v2f a1f[8];
#pragma unroll
    for (int ks = 0; ks < 8; ++ks) a1f[ks] = *(const v2f*)(xr + 4 * ks);

    v8f acc[8];
#pragma unroll
    for (int nt = 0; nt < 8; ++nt) { v8f z = {}; acc[nt] = z; }
#pragma unroll
    for (int nt = 0; nt < 8; ++nt) {
#pragma unroll
      for (int ks = 0; ks < 8; ++ks) {
        const int k = 4 * ks + 2 * hi;
        v2f bf;
        bf.x = sW1[(k    ) * HID + nt * 16 + ln];
        bf.y = sW1[(k + 1) * HID + nt * 16 + ln];
        acc[nt] = wmma_k4(a1f[ks], bf, acc[nt]);
      }
    }

    // ---- bias + sigmoid, D-layout -> LDS (h tile, row-major 16x128) ----
    {
      float* hp = hb + (8 * hi) * HID + ln;
#pragma unroll
      for (int nt = 0; nt < 8; ++nt) {
#pragma unroll
        for (int j = 0; j < 8; ++j) {
          float v = acc[nt][j] + c1f[nt];
          v = 1.0f / (1.0f + __expf(-v));
          hp[j * HID + nt * 16] = v;
        }
      }
    }
    wait_ds0();  // DS ops are per-wave in-order; LDS now holds full h tile

    // ---- GEMM2: A = h tile (16x128) read back in A-layout, B = W2 ----
    const float* hr = hb + ln * HID + 2 * hi;
    v2f a2f[32];
#pragma unroll
    for (int ks = 0; ks < 32; ++ks)
      a2f[ks] = *(const v2f*)(hr + 4 * ks);

    v8f acc2[2];
    { v8f z = {}; acc2[0] = z; acc2[1] = z; }
#pragma unroll
    for (int nt = 0; nt < 2; ++nt) {
#pragma unroll
      for (int ks = 0; ks < 32; ++ks) {
        const int k = 4 * ks + 2 * hi;
        v2f bf;
        bf.x = sW2[(k    ) * ENC + nt * 16 + ln];
        bf.y = sW2[(k + 1) * ENC + nt * 16 + ln];
        acc2[nt] = wmma_k4(a2f[ks], bf, acc2[nt]);
      }
    }

    // D tile store: one base pointer, constant row offsets.
    float* p = buf + (size_t)(row0 + 8 * hi) * ENC + ln;
#pragma unroll
    for (int j = 0; j < 8; ++j) {
      p[j * ENC]      = acc2[0][j] + c2f[0];
      p[j * ENC + 16] = acc2[1][j] + c2f[1];
    }
    wait_ds0();  // retire a2f LDS reads before next iteration rewrites hb
  }
}

// ---------------------------------------------------------------------------
// Kernel 4: FIR2 (xf ⊛ b2, K=288) and FIR3 (u ⊛ b3, K=36), N=4 padded to 16.
// ---------------------------------------------------------------------------
__global__ void __launch_bounds__(256) k_fir23(const float* __restrict__ xf,
                                               const float* __restrict__ u,
                                               const float* __restrict__ b2,
                                               const float* __restrict__ b3,
                                               float* __restrict__ bu2,
                                               float* __restrict__ bu3) {
  __shared__ float sB2[NBT * ENC * OUTC];  // 1152
  __shared__ float sB3[NBT * CIN * OUTC];  // 144
  for (int i = threadIdx.x; i < NBT * ENC * OUTC; i += blockDim.x) sB2[i] = b2[i];
  for (int i = threadIdx.x; i < NBT * CIN * OUTC; i += blockDim.x) sB3[i] = b3[i];
  __syncthreads();

  const int lane = threadIdx.x & 31, ln = lane & 15, hi = lane >> 4;
  const int wid = blockIdx.x * (blockDim.x >> 5) + (threadIdx.x >> 5);
  const int nw  = gridDim.x * (blockDim.x >> 5);

  // Hoisted tile-invariant FIR3 B fragments (9 x v2f), N padded: cols >= 4 are 0.
  v2f bf3[NBT];
#pragma unroll
  for (int tap = 0; tap < NBT; ++tap) {
    const int k = tap * CIN + 2 * hi;
    bf3[tap].x = (ln < OUTC) ? sB3[(k    ) * OUTC + ln] : 0.f;
    bf3[tap].y = (ln < OUTC) ? sB3[(k + 1) * OUTC + ln] : 0.f;
  }
  wait_ds0();

  for (int tile = wid; tile < NTIL; tile += nw) {
    const int row0 = tile * 16;
    const int b = row0 / T, t0 = row0 % T;
    const bool interior = (t0 != 0);  // wave-uniform

    // ---- FIR2: K index = tap*32 + c (c even within step, never crosses a tap)
    v8f acc2 = {};
    if (interior) {
      const float* xb = xf + ((size_t)b * T + t0 + ln) * ENC;
#pragma unroll
      for (int ks = 0; ks < 72; ++ks) {
        const int k = 4 * ks + 2 * hi;
        const int tap = k >> 5, c = k & 31;
        v2f a = *(const v2f*)(xb - tap * ENC + c);
        v2f bf;
        bf.x = (ln < OUTC) ? sB2[(k    ) * OUTC + ln] : 0.f;
        bf.y = (ln < OUTC) ? sB2[(k + 1) * OUTC + ln] : 0.f;
        acc2 = wmma_k4(a, bf, acc2);
      }
    } else {
#pragma unroll
      for (int ks = 0; ks < 72; ++ks) {
        const int k = 4 * ks + 2 * hi;
        const int tap = k >> 5, c = k & 31;
        const int t = ln - tap;
        v2f a = {};
        if (t >= 0) a = *(const v2f*)(xf + ((size_t)b * T + t) * ENC + c);
        v2f bf;
        bf.x = (ln < OUTC) ? sB2[(k    ) * OUTC + ln] : 0.f;
        bf.y = (ln < OUTC) ? sB2[(k + 1) * OUTC + ln] : 0.f;
        acc2 = wmma_k4(a, bf, acc2);
      }
    }

    // ---- FIR3: K index = tap*4 + c
    v8f acc3 = {};
    if (interior) {
      const float* ub = u + ((size_t)b * T + t0 + ln) * CIN + 2 * hi;
#pragma unroll
      for (int tap = 0; tap < NBT; ++tap)
        acc3 = wmma_k4(*(const v2f*)(ub - tap * CIN), bf3[tap], acc3);
    } else {
#pragma unroll
      for (int tap = 0; tap < NBT; ++tap) {
        const int t = ln - tap;
        v2f a = {};
        if (t >= 0) a = *(const v2f*)(u + ((size_t)b * T + t) * CIN + 2 * hi);
        acc3 = wmma_k4(a, bf3[tap], acc3);
      }
    }

    if (ln < OUTC) {
      float* p2 = bu2 + (size_t)(row0 + 8 * hi) * OUTC + ln;
      float* p3 = bu3 + (size_t)(row0 + 8 * hi) * OUTC + ln;
#pragma unroll
      for (int j = 0; j < 8; ++j) {
        p2[j * OUTC] = acc2[j];
        p3[j * OUTC] = acc3[j];
      }
    }
  }
}

// ---------------------------------------------------------------------------
// Kernel 5: IIR2 + IIR3 (x0 initial state for both) and final sum -> out.
// y[-k] = x0[b, NA-k, o]; y[t] = v[t] + sum_n a[n]*y[t-1-n].
// ---------------------------------------------------------------------------
__global__ void k_iir23(const float* __restrict__ bu2v,
                        const float* __restrict__ bu3v,
                        const float* __restrict__ x0,
                        const float* __restrict__ a2,
                        const float* __restrict__ a3,
                        float* __restrict__ out) {
  const int tid = threadIdx.x;
  if (tid >= BATCH * OUTC) return;
  const int b = tid / OUTC, o = tid % OUTC;
  const float p0 = a2[0 * OUTC + o], p1 = a2[1 * OUTC + o];
  const float p2 = a2[2 * OUTC + o], p3 = a2[3 * OUTC + o];
  const float q0 = a3[0 * OUTC + o], q1 = a3[1 * OUTC + o];
  const float q2 = a3[2 * OUTC + o], q3 = a3[3 * OUTC + o];
  float y1 = x0[((size_t)b * NA + 3) * OUTC + o];
  float y2 = x0[((size_t)b * NA + 2) * OUTC + o];
  float y3 = x0[((size_t)b * NA + 1) * OUTC + o];
  float y4 = x0[((size_t)b * NA + 0) * OUTC + o];
  float z1 = y1, z2 = y2, z3 = y3, z4 = y4;
  const float* pa = bu2v + (size_t)b * T * OUTC + o;
  const float* pb = bu3v + (size_t)b * T * OUTC + o;
  float*       po = out  + (size_t)b * T * OUTC + o;
  for (int t = 0; t < T; ++t) {
    float v2 = *pa;
    v2 = fmaf(p0, y1, fmaf(p1, y2, fmaf(p2, y3, fmaf(p3, y4, v2))));
    float v3 = *pb;
    v3 = fmaf(q0, z1, fmaf(q1, z2, fmaf(q2, z3, fmaf(q3, z4, v3))));
    *po = v2 + v3;
    pa += OUTC; pb += OUTC; po += OUTC;
    y4 = y3; y3 = y2; y2 = y1; y1 = v2;
    z4 = z3; z3 = z2; z2 = z1; z1 = v3;
  }
}

// ---------------------------------------------------------------------------
extern "C" void kernel_launch(void* const* d_in, const int* in_sizes, int n_in,
                              void* d_out, int out_size, void* d_ws, size_t ws_size,
                              hipStream_t stream) {
  const float* u  = (const float*)d_in[0];
  const float* x0 = (const float*)d_in[1];
  const float* b1 = (const float*)d_in[2];
  const float* a1 = (const float*)d_in[3];
  const float* W1 = (const float*)d_in[4];
  const float* c1 = (const float*)d_in[5];
  const float* W2 = (const float*)d_in[6];
  const float* c2 = (const float*)d_in[7];
  const float* b2 = (const float*)d_in[8];
  const float* a2 = (const float*)d_in[9];
  const float* b3 = (const float*)d_in[10];
  const float* a3 = (const float*)d_in[11];
  float* out = (float*)d_out;

  char* ws = (char*)d_ws;
  float* buf  = (float*)ws;                                         // ROWS*ENC (bu1 -> x -> xf, in-place)
  float* bu2b = (float*)(ws + (size_t)ROWS * ENC * sizeof(float));  // ROWS*OUTC
  float* bu3b = (float*)(ws + ((size_t)ROWS * ENC + (size_t)ROWS * OUTC) * sizeof(float));

  k_fir1 <<<256, 256, 0, stream>>>(u, b1, buf);
  k_iir_enc<<<2, 256, 0, stream>>>(buf, a1);
  k_mlp  <<<256, 128, 0, stream>>>(buf, W1, c1, W2, c2);
  k_fir23<<<256, 256, 0, stream>>>(buf, u, b2, b3, bu2b, bu3b);
  k_iir23<<<1, 64, 0, stream>>>(bu2b, bu3b, x0, a2, a3, out);
}